// CrystalGraphConv_15865609191625
// MI455X (gfx1250) — compile-verified
//
#include <hip/hip_runtime.h>
#include <hip/hip_bf16.h>

typedef float v2f __attribute__((ext_vector_type(2)));
typedef float v8f __attribute__((ext_vector_type(8)));

#define D 64  // D_IN == D_OUT == 64

// ---------------------------------------------------------------------------
// Kernel 1: x_t = x @ weight  via V_WMMA_F32_16X16X4_F32 (exact fp32).
// One wave computes one 16x16 tile of x_t; 8 waves per block cover
// 2 M-tiles x 4 N-tiles. K=64 in 16 steps of 4.
// ---------------------------------------------------------------------------
__global__ __launch_bounds__(256)
void cgc_xform_wmma(const float* __restrict__ x, const float* __restrict__ w,
                    float* __restrict__ xt, int n_nodes) {
  const int lane = threadIdx.x & 31;
  const int wave = threadIdx.x >> 5;              // 0..7
  const int mtile = blockIdx.x * 2 + (wave >> 2); // 2 M-tiles per block
  const int ntile = wave & 3;                     // 4 N-tiles (D_OUT = 64)
  const int m0 = mtile * 16;
  const int n0 = ntile * 16;
  if (m0 >= n_nodes) return;                      // wave-uniform: EXEC stays all-1

  const int row = lane & 15;                      // M (for A) / N (for B)
  const int hi  = lane >> 4;                      // K-half selector

  v8f c = {};
#pragma unroll
  for (int k = 0; k < D; k += 4) {
    // A 16x4 f32: lanes 0-15 hold K={0,1}, lanes 16-31 hold K={2,3} for row M
    const float* ap = x + (size_t)(m0 + row) * D + (k + hi * 2);
    v2f a; a.x = ap[0]; a.y = ap[1];              // 8B-aligned -> b64 load
    // B 4x16 f32: lanes 0-15 hold K={0,1}, lanes 16-31 hold K={2,3} for col N
    const float* bp = w + (size_t)(k + hi * 2) * D + (n0 + row);
    v2f b; b.x = bp[0]; b.y = bp[D];
    c = __builtin_amdgcn_wmma_f32_16x16x4_f32(false, a, false, b,
                                              (short)0, c, false, false);
  }
  // C/D 16x16 f32 layout: VGPR v, lanes 0-15 -> M=v, lanes 16-31 -> M=v+8
#pragma unroll
  for (int v = 0; v < 8; ++v) {
    const int m = m0 + v + hi * 8;
    xt[(size_t)m * D + n0 + row] = c[v];
  }
}

// ---------------------------------------------------------------------------
// Kernel 2: out[n, d] = bias[d]   (float4 vectorized init before scatter-add)
// ---------------------------------------------------------------------------
__global__ __launch_bounds__(256)
void cgc_init_out(float* __restrict__ out, const float* __restrict__ bias,
                  int n4 /* total float4 count */) {
  const int t = blockIdx.x * blockDim.x + threadIdx.x;
  if (t >= n4) return;
  const int d4 = (t & 15) << 2;                   // dim group within row
  const float4 bv = *reinterpret_cast<const float4*>(bias + d4);
  reinterpret_cast<float4*>(out)[t] = bv;
}

// ---------------------------------------------------------------------------
// Kernel 3: per edge e:  out[dst[e]] += x_t[src[e]] * sigmoid(a_e*W + b)
// 16 lanes per edge, 4 dims (one float4) per lane: 256B coalesced gather,
// 4 hardware global_atomic_add_f32 per lane into L2-resident `out`.
// ---------------------------------------------------------------------------
__global__ __launch_bounds__(256)
void cgc_edge_scatter(const float* __restrict__ xt,
                      const long long* __restrict__ eidx,  // [2, E] int64
                      const float* __restrict__ eattr,     // [E, 1]
                      const float* __restrict__ ewW,       // [1, 64]
                      const float* __restrict__ ewb,       // [64]
                      float* __restrict__ out, int n_edges) {
  const int t = blockIdx.x * blockDim.x + threadIdx.x;
  const int e  = t >> 4;
  const int d4 = (t & 15) << 2;                   // dims d4 .. d4+3
  if (e >= n_edges) return;

  const long long s = eidx[e];                    // message source
  const long long d = eidx[(size_t)n_edges + e];  // message destination
  const float a = eattr[e];

  const float4 Wv = *reinterpret_cast<const float4*>(ewW + d4);
  const float4 Bv = *reinterpret_cast<const float4*>(ewb + d4);
  const float4 Xv = *reinterpret_cast<const float4*>(xt + (size_t)s * D + d4);

  const float g0 = 1.0f / (1.0f + __expf(-(a * Wv.x + Bv.x)));
  const float g1 = 1.0f / (1.0f + __expf(-(a * Wv.y + Bv.y)));
  const float g2 = 1.0f / (1.0f + __expf(-(a * Wv.z + Bv.z)));
  const float g3 = 1.0f / (1.0f + __expf(-(a * Wv.w + Bv.w)));

  float* op = out + (size_t)d * D + d4;
  unsafeAtomicAdd(op + 0, Xv.x * g0);
  unsafeAtomicAdd(op + 1, Xv.y * g1);
  unsafeAtomicAdd(op + 2, Xv.z * g2);
  unsafeAtomicAdd(op + 3, Xv.w * g3);
}

// ---------------------------------------------------------------------------
// Launch
// inputs: 0:x [N,64] f32, 1:edge_index [2,E] i64, 2:edge_attr [E,1] f32,
//         3:weight [64,64] f32, 4:ew_W [1,64] f32, 5:ew_b [64] f32, 6:bias [64] f32
// ---------------------------------------------------------------------------
extern "C" void kernel_launch(void* const* d_in, const int* in_sizes, int n_in,
                              void* d_out, int out_size, void* d_ws, size_t ws_size,
                              hipStream_t stream) {
  const float*     x      = (const float*)d_in[0];
  const long long* eidx   = (const long long*)d_in[1];
  const float*     eattr  = (const float*)d_in[2];
  const float*     weight = (const float*)d_in[3];
  const float*     ewW    = (const float*)d_in[4];
  const float*     ewb    = (const float*)d_in[5];
  const float*     bias   = (const float*)d_in[6];
  float*           out    = (float*)d_out;
  float*           xt     = (float*)d_ws;          // [N, 64] fp32 scratch

  const int n_nodes = in_sizes[0] / D;             // 100000
  const int n_edges = in_sizes[2];                 // 1600000 (EDGE_DIM == 1)

  // 1) node transform (WMMA fp32)
  {
    const int mtiles = (n_nodes + 15) / 16;
    const int blocks = (mtiles + 1) / 2;           // 2 M-tiles per 256-thread block
    cgc_xform_wmma<<<blocks, 256, 0, stream>>>(x, weight, xt, n_nodes);
  }
  // 2) out = bias (broadcast)
  {
    const int n4 = n_nodes * (D / 4);
    cgc_init_out<<<(n4 + 255) / 256, 256, 0, stream>>>(out, bias, n4);
  }
  // 3) gather * sigmoid -> scatter-add
  {
    const long long threads = (long long)n_edges * 16;
    const int blocks = (int)((threads + 255) / 256);
    cgc_edge_scatter<<<blocks, 256, 0, stream>>>(xt, eidx, eattr, ewW, ewb,
                                                 out, n_edges);
  }
}